// NonLinearActorNet_29961691857572
// MI455X (gfx1250) — compile-verified
//
#include <hip/hip_runtime.h>
#include <math.h>

#define ABS_DIM    512
#define HIDDEN     512
#define ACTION_DIM 18
#define N_TASKS    16
#define M_TILE     32   // rows per workgroup tile
#define KP         16   // K-panel depth staged by TDM

typedef __attribute__((ext_vector_type(2))) float v2f;
typedef __attribute__((ext_vector_type(8))) float v8f;
typedef unsigned int u32;
typedef __attribute__((ext_vector_type(4))) u32 u32x4;
typedef __attribute__((ext_vector_type(8))) int i32x8;
typedef __attribute__((ext_vector_type(4))) int i32x4;

// ---------------- grouping prep kernels ----------------

__global__ void zero_ws_kernel(int* ws) {
    int t = threadIdx.x;
    if (t < 48) ws[t] = 0;   // counts[16], offsets[16], cursors[16]
}

__global__ void hist_kernel(const int* __restrict__ task_id, int* __restrict__ counts, int n) {
    int i = blockIdx.x * blockDim.x + threadIdx.x;
    if (i < n) atomicAdd(&counts[task_id[i]], 1);
}

__global__ void scan_kernel(int* ws) {
    if (threadIdx.x == 0) {
        int acc = 0;
        for (int t = 0; t < N_TASKS; ++t) { ws[16 + t] = acc; acc += ws[t]; }
    }
}

__global__ void scatter_kernel(const int* __restrict__ task_id, int* __restrict__ ws,
                               int* __restrict__ perm, int n) {
    int i = blockIdx.x * blockDim.x + threadIdx.x;
    if (i < n) {
        int t = task_id[i];
        int pos = ws[16 + t] + atomicAdd(&ws[32 + t], 1);
        perm[pos] = i;
    }
}

// ---------------- TDM helper: DMA a KP x 512 fp32 panel of W1 into LDS ----------------
// D# per CDNA5 ISA ch.8: group0 {count/lds_addr/global_addr/type},
// group1 {data_size, tensor dims, tile dims, strides}. 2D tensor -> groups 2/3 zero.

__device__ __forceinline__ void tdm_load_panel(const float* gsrc, float* lds_dst) {
    unsigned long long ga = (unsigned long long)(uintptr_t)gsrc;
    u32x4 g0;
    g0.x = 1u;                                            // count=1, user mode, no gather
    g0.y = (u32)(uintptr_t)lds_dst;                       // LDS byte offset (addr[31:0])
    g0.z = (u32)ga;                                       // global_addr[31:0]
    g0.w = (u32)((ga >> 32) & 0x01FFFFFFu) | 0x80000000u; // global_addr[56:32] | type=2
    i32x8 g1;
    g1[0] = (int)(2u << 16);                  // workgroup_mask=0, data_size=2 (4 bytes)
    g1[1] = (int)(512u << 16);                // tensor_dim0[15:0] at [31:16]
    g1[2] = (int)(512u << 16);                // tensor_dim0 hi=0 | tensor_dim1[15:0]
    g1[3] = (int)(512u << 16);                // tensor_dim1 hi=0 | tile_dim0=512
    g1[4] = KP;                               // tile_dim1=KP, tile_dim2=0
    g1[5] = 512;                              // tensor_dim0_stride low 32 (elements)
    g1[6] = 0;                                // stride0 hi | stride1 lo
    g1[7] = 0;                                // stride1 hi
    i32x4 gz = {0, 0, 0, 0};
#if defined(__clang_major__) && __clang_major__ >= 23
    i32x8 gz8 = {0, 0, 0, 0, 0, 0, 0, 0};
    __builtin_amdgcn_tensor_load_to_lds(g0, g1, gz, gz, gz8, 0);
#else
    __builtin_amdgcn_tensor_load_to_lds(g0, g1, gz, gz, 0);
#endif
}

// ---------------- fused grouped-GEMM actor kernel ----------------
// grid = (maxTiles, N_TASKS), block = 256 (8 waves)

__global__ __launch_bounds__(256)
void actor_fused_kernel(const float* __restrict__ x,
                        const int*   __restrict__ action,
                        const float* __restrict__ W1, const float* __restrict__ b1,
                        const float* __restrict__ W2, const float* __restrict__ b2,
                        const int*   __restrict__ ws, const int* __restrict__ perm,
                        float* __restrict__ out)
{
    const int t   = blockIdx.y;
    const int cnt = ws[t];          // samples of this task
    const int off = ws[16 + t];     // start in perm[]
    const int tiles = (cnt + M_TILE - 1) / M_TILE;
    if ((int)blockIdx.x >= tiles) return;
    const int row0 = blockIdx.x * M_TILE;

    extern __shared__ float smem[];
    float* As = smem;                         // [32][512]   64 KB
    float* Hs = As + M_TILE * ABS_DIM;        // [32][512]   64 KB
    float* Bp = Hs + M_TILE * HIDDEN;         // [2][KP][512] 64 KB (TDM double buffer)
    float* Lg = Bp + 2 * KP * HIDDEN;         // [32][32]     4 KB
    int*   rI = (int*)(Lg + M_TILE * 32);     // [32]

    const int tid = threadIdx.x;

    // resolve original sample indices for this tile
    if (tid < M_TILE) {
        int gp = row0 + tid;
        rI[tid] = (gp < cnt) ? perm[off + gp] : -1;
    }
    __syncthreads();

    // gather A tile (zero padded rows), float4 per thread
    for (int idx = tid; idx < M_TILE * (ABS_DIM / 4); idx += 256) {
        int r = idx >> 7;                 // ABS_DIM/4 == 128
        int c = (idx & 127) << 2;
        int src = rI[r];
        float4 v = make_float4(0.f, 0.f, 0.f, 0.f);
        if (src >= 0) v = *(const float4*)(x + (size_t)src * ABS_DIM + c);
        *(float4*)(As + r * ABS_DIM + c) = v;
    }

    const int wave = tid >> 5;
    const int lane = tid & 31;
    const int half = lane >> 4;   // 0: lanes 0-15, 1: lanes 16-31
    const int l16  = lane & 15;
    const v8f vzero = {};

    const float* Wt = W1 + (size_t)t * ABS_DIM * HIDDEN;
    const int NPANELS = ABS_DIM / KP;   // 32

    // preload panel 0 via TDM (wave 0 only; DMA ignores EXEC, tracked by TENSORcnt)
    if (wave == 0) {
        tdm_load_panel(Wt, Bp);
        __builtin_amdgcn_s_wait_tensorcnt(0);
    }
    __syncthreads();   // also covers the As gather above

    // ---- Layer 1: H = relu(A * W1[t] + b1[t]),  A:32x512, W1:512x512 ----
    {
        const int mbase = (wave >> 2) * 16;   // which 16-row half
        const int n0    = (wave & 3) * 128;   // 128-col stripe

        v8f acc[8];
        #pragma unroll
        for (int j = 0; j < 8; ++j) acc[j] = vzero;

        const float* Arow = As + (size_t)(mbase + l16) * ABS_DIM;

        for (int p = 0; p < NPANELS; ++p) {
            // kick off DMA of next panel while computing on current
            if (wave == 0 && (p + 1) < NPANELS)
                tdm_load_panel(Wt + (size_t)(p + 1) * KP * HIDDEN,
                               Bp + ((p + 1) & 1) * KP * HIDDEN);

            const float* Bcur = Bp + (p & 1) * KP * HIDDEN;
            #pragma unroll
            for (int kk = 0; kk < KP; kk += 4) {
                // A frag 16x4: VGPR0 = K (k + 0/2), VGPR1 = K (k + 1/3)
                v2f a;
                a.x = Arow[p * KP + kk + 2 * half];
                a.y = Arow[p * KP + kk + 2 * half + 1];
                const float* Bk0 = Bcur + (kk + 2 * half) * HIDDEN;
                const float* Bk1 = Bk0 + HIDDEN;
                #pragma unroll
                for (int j = 0; j < 8; ++j) {
                    int n = n0 + j * 16 + l16;
                    v2f b;
                    b.x = Bk0[n];
                    b.y = Bk1[n];
                    acc[j] = __builtin_amdgcn_wmma_f32_16x16x4_f32(
                        false, a, false, b, (short)0, acc[j], false, false);
                }
            }

            if (wave == 0) __builtin_amdgcn_s_wait_tensorcnt(0);
            __syncthreads();   // publish next panel / protect current from overwrite
        }

        // epilogue: bias + relu -> Hs
        #pragma unroll
        for (int j = 0; j < 8; ++j) {
            int col = n0 + j * 16 + l16;
            float bias = b1[t * HIDDEN + col];
            #pragma unroll
            for (int v = 0; v < 8; ++v) {
                int row = mbase + v + 8 * half;   // C/D: VGPR v -> M = v (+8 for hi lanes)
                float val = acc[j][v] + bias;
                Hs[row * HIDDEN + col] = val > 0.f ? val : 0.f;
            }
        }
    }
    __syncthreads();

    // ---- Layer 2: logits[32 x 32(pad 18)] = H * W2[t] + b2[t], waves 0..3 ----
    if (wave < 4) {
        const int mbase = (wave >> 1) * 16;
        const int nsub  = (wave & 1) * 16;
        const float* W2t = W2 + (size_t)t * HIDDEN * ACTION_DIM;
        v8f acc = vzero;
        const float* Hrow = Hs + (size_t)(mbase + l16) * HIDDEN;
        const int col = nsub + l16;
        const bool cv = col < ACTION_DIM;
        for (int k = 0; k < HIDDEN; k += 4) {
            v2f a;
            a.x = Hrow[k + 2 * half];
            a.y = Hrow[k + 2 * half + 1];
            v2f b;
            b.x = cv ? W2t[(size_t)(k + 2 * half) * ACTION_DIM + col] : 0.f;
            b.y = cv ? W2t[(size_t)(k + 2 * half + 1) * ACTION_DIM + col] : 0.f;
            acc = __builtin_amdgcn_wmma_f32_16x16x4_f32(
                false, a, false, b, (short)0, acc, false, false);
        }
        float bias = cv ? b2[t * ACTION_DIM + col] : 0.f;
        #pragma unroll
        for (int v = 0; v < 8; ++v) {
            int row = mbase + v + 8 * half;
            Lg[row * 32 + col] = acc[v] + bias;
        }
    }
    __syncthreads();

    // ---- log-softmax, log pi(a), entropy: one lane per row ----
    if (tid < M_TILE) {
        int src = rI[tid];
        if (src >= 0) {
            const float* lr = Lg + tid * 32;
            float m = lr[0];
            #pragma unroll
            for (int c = 1; c < ACTION_DIM; ++c) m = fmaxf(m, lr[c]);
            float se = 0.f;
            #pragma unroll
            for (int c = 0; c < ACTION_DIM; ++c) se += __expf(lr[c] - m);
            float logZ = m + __logf(se);
            int a = action[src];
            float lp = lr[a] - logZ;
            float ent = 0.f;
            #pragma unroll
            for (int c = 0; c < ACTION_DIM; ++c) {
                float ls = lr[c] - logZ;
                ent -= __expf(ls) * ls;
            }
            out[(size_t)src * 2]     = lp;
            out[(size_t)src * 2 + 1] = ent;
        }
    }
}

// ---------------- launcher ----------------

extern "C" void kernel_launch(void* const* d_in, const int* in_sizes, int n_in,
                              void* d_out, int out_size, void* d_ws, size_t ws_size,
                              hipStream_t stream) {
    const float* x       = (const float*)d_in[0];
    const int*   task_id = (const int*)  d_in[1];
    const int*   action  = (const int*)  d_in[2];
    const float* W1      = (const float*)d_in[3];
    const float* b1      = (const float*)d_in[4];
    const float* W2      = (const float*)d_in[5];
    const float* b2      = (const float*)d_in[6];
    float* out = (float*)d_out;

    int* ws   = (int*)d_ws;   // [0..15] counts, [16..31] offsets, [32..47] cursors
    int* perm = ws + 48;
    const int B = in_sizes[1];   // number of samples (task_id count)

    zero_ws_kernel<<<1, 64, 0, stream>>>(ws);
    hist_kernel<<<(B + 255) / 256, 256, 0, stream>>>(task_id, ws, B);
    scan_kernel<<<1, 32, 0, stream>>>(ws);
    scatter_kernel<<<(B + 255) / 256, 256, 0, stream>>>(task_id, ws, perm, B);

    size_t smem = (size_t)(M_TILE * ABS_DIM       // As
                         + M_TILE * HIDDEN        // Hs
                         + 2 * KP * HIDDEN        // TDM panels
                         + M_TILE * 32) * sizeof(float)
                + M_TILE * sizeof(int);
    dim3 grid((B + M_TILE - 1) / M_TILE, N_TASKS);
    actor_fused_kernel<<<grid, 256, smem, stream>>>(x, action, W1, b1, W2, b2,
                                                    ws, perm, out);
}